// HeteroRGCN_90520730730826
// MI455X (gfx1250) — compile-verified
//
#include <hip/hip_runtime.h>
#include <math.h>

#define D 128

typedef __attribute__((ext_vector_type(2))) float v2f;
typedef __attribute__((ext_vector_type(8))) float v8f;

// ---------------------------------------------------------------------------
// Init: zero outputs + per-dst degree counters
// ---------------------------------------------------------------------------
__global__ __launch_bounds__(256) void hrgcn_init_kernel(
    float* __restrict__ out, int nOut,
    unsigned int* __restrict__ cnt, int nC) {
  int i = blockIdx.x * blockDim.x + threadIdx.x;
  int stride = gridDim.x * blockDim.x;
  for (int j = i; j < nOut; j += stride) out[j] = 0.f;
  for (int j = i; j < nC;   j += stride) cnt[j] = 0u;
}

// ---------------------------------------------------------------------------
// GEMM: Y[n,128] = X[n,128] @ W[128,128] + b, via V_WMMA_F32_16X16X4_F32.
// Block = 256 threads = 8 waves; each wave owns one 16x16 N-tile of a 16-row
// M-tile; K=128 in steps of 4. A-frag: lanes 0-15 hold K={k0,k0+1}, lanes
// 16-31 K={k0+2,k0+3}. B-frag mirrors with N striped across lanes. C: VGPR g
// holds rows g (lanes 0-15) and g+8 (lanes 16-31), N = lane%16.
// ---------------------------------------------------------------------------
__global__ __launch_bounds__(256) void hrgcn_gemm_bias_kernel(
    const float* __restrict__ X, const float* __restrict__ W,
    const float* __restrict__ b, float* __restrict__ Y, int n) {
  const int wave  = threadIdx.x >> 5;        // 0..7  -> N tile
  const int lane  = threadIdx.x & 31;
  const int m     = lane & 15;
  const int half  = lane >> 4;               // 0 or 1
  const int rowBase = blockIdx.x * 16;
  const int nBase   = wave * 16;

  const float* Xrow = X + (size_t)(rowBase + m) * D;
  v8f c = {};
  #pragma unroll
  for (int k0 = 0; k0 < D; k0 += 4) {
    const int ka = k0 + 2 * half;
    v2f a;  a.x  = Xrow[ka];              a.y  = Xrow[ka + 1];
    v2f bb; bb.x = W[ka * D + nBase + m]; bb.y = W[(ka + 1) * D + nBase + m];
    c = __builtin_amdgcn_wmma_f32_16x16x4_f32(
        /*neg_a=*/false, a, /*neg_b=*/false, bb,
        /*c_mod=*/(short)0, c, /*reuse_a=*/false, /*reuse_b=*/false);
  }
  const float bias = b[nBase + m];
  #pragma unroll
  for (int g = 0; g < 8; ++g) {
    const int row = rowBase + g + 8 * half;
    if (row < n) Y[(size_t)row * D + nBase + m] = c[g] + bias;
  }
}

// ---------------------------------------------------------------------------
// Per-node scalar: out[node] = dot(Wh[node,:], a[0:128]); one wave per node.
// ---------------------------------------------------------------------------
__global__ __launch_bounds__(256) void hrgcn_dot_a_kernel(
    const float* __restrict__ Wh, const float* __restrict__ a,
    float* __restrict__ out, int n) {
  const int node = blockIdx.x * (blockDim.x >> 5) + (threadIdx.x >> 5);
  const int lane = threadIdx.x & 31;
  if (node >= n) return;
  const float* row = Wh + (size_t)node * D;
  float sum = 0.f;
  #pragma unroll
  for (int f = 0; f < 4; ++f) sum += row[lane + 32 * f] * a[lane + 32 * f];
  #pragma unroll
  for (int off = 16; off > 0; off >>= 1) sum += __shfl_xor(sum, off, 32);
  if (lane == 0) out[node] = sum;
}

// ---------------------------------------------------------------------------
// CSR build 1/3: histogram of destination degrees
// ---------------------------------------------------------------------------
__global__ __launch_bounds__(256) void hrgcn_hist_kernel(
    const int* __restrict__ dst, unsigned int* __restrict__ cnt, int E_) {
  const int i = blockIdx.x * blockDim.x + threadIdx.x;
  if (i < E_) atomicAdd(&cnt[dst[i]], 1u);
}

// ---------------------------------------------------------------------------
// CSR build 2/3: single-block exclusive prefix scan over n counters.
// Each thread owns a contiguous chunk; Hillis-Steele scan over chunk sums.
// Writes rowptr[0..n] and a cursor copy for the fill pass.
// ---------------------------------------------------------------------------
__global__ __launch_bounds__(1024) void hrgcn_scan_kernel(
    const unsigned int* __restrict__ cnt, unsigned int* __restrict__ rowptr,
    unsigned int* __restrict__ cursor, int n) {
  __shared__ unsigned int partial[1024];
  const int t = threadIdx.x;
  const int chunk = (n + 1023) / 1024;
  const int beg = t * chunk;
  const int end = (beg + chunk < n) ? (beg + chunk) : n;
  unsigned int local = 0;
  for (int i = beg; i < end; ++i) local += cnt[i];
  partial[t] = local;
  __syncthreads();
  for (int off = 1; off < 1024; off <<= 1) {
    unsigned int v = (t >= off) ? partial[t - off] : 0u;
    __syncthreads();
    partial[t] += v;
    __syncthreads();
  }
  unsigned int run = (t == 0) ? 0u : partial[t - 1];
  for (int i = beg; i < end; ++i) {
    rowptr[i] = run;
    cursor[i] = run;
    run += cnt[i];
  }
  if (t == 0) rowptr[n] = partial[1023];
}

// ---------------------------------------------------------------------------
// CSR build 3/3: scatter source ids into dst-grouped order
// ---------------------------------------------------------------------------
__global__ __launch_bounds__(256) void hrgcn_fill_kernel(
    const int* __restrict__ src, const int* __restrict__ dst,
    unsigned int* __restrict__ cursor, int* __restrict__ srcs, int E_) {
  const int i = blockIdx.x * blockDim.x + threadIdx.x;
  if (i >= E_) return;
  const unsigned int pos = atomicAdd(&cursor[dst[i]], 1u);
  srcs[pos] = src[i];
}

// ---------------------------------------------------------------------------
// Aggregate: one wave per destination node. Wave-reduce softmax max & sum of
// the cheap scalar logits, then sequentially gather each source's 512B Wh row
// (float4/lane) and accumulate the 128-wide result in registers. Atomic-free:
// out[d,:] += acc (stream-ordered kernels give the cross-etype sum).
// ---------------------------------------------------------------------------
__global__ __launch_bounds__(256) void hrgcn_agg_kernel(
    const unsigned int* __restrict__ rowptr, const int* __restrict__ srcs,
    const float* __restrict__ el, const float* __restrict__ er,
    const float* __restrict__ Wh, float* __restrict__ out, int n) {
  const int d    = blockIdx.x * (blockDim.x >> 5) + (threadIdx.x >> 5);
  const int lane = threadIdx.x & 31;
  if (d >= n) return;
  const unsigned int beg = rowptr[d];
  const int deg = (int)(rowptr[d + 1] - beg);
  if (deg == 0) return;
  const float erd = er[d];

  // pass A: segment max of leaky_relu(el[src] + er[d])
  float m = -INFINITY;
  for (int j = lane; j < deg; j += 32) {
    float x = el[srcs[beg + j]] + erd;
    x = x > 0.f ? x : 0.01f * x;
    m = fmaxf(m, x);
  }
  #pragma unroll
  for (int off = 16; off > 0; off >>= 1) m = fmaxf(m, __shfl_xor(m, off, 32));

  // pass B: sum of exp(e - m)
  float ssum = 0.f;
  for (int j = lane; j < deg; j += 32) {
    float x = el[srcs[beg + j]] + erd;
    x = x > 0.f ? x : 0.01f * x;
    ssum += __expf(x - m);
  }
  #pragma unroll
  for (int off = 16; off > 0; off >>= 1) ssum += __shfl_xor(ssum, off, 32);
  const float inv = 1.0f / ssum;

  // pass C: weighted feature accumulation in registers
  float4 acc = {0.f, 0.f, 0.f, 0.f};
  for (int j = 0; j < deg; ++j) {
    const int s = srcs[beg + j];
    float x = el[s] + erd;
    x = x > 0.f ? x : 0.01f * x;
    const float alpha = __expf(x - m) * inv;
    const float4 v = ((const float4*)(Wh + (size_t)s * D))[lane];
    acc.x += alpha * v.x; acc.y += alpha * v.y;
    acc.z += alpha * v.z; acc.w += alpha * v.w;
  }
  float* orow = out + (size_t)d * D + lane * 4;
  float4 o = *((float4*)orow);
  o.x += acc.x; o.y += acc.y; o.z += acc.z; o.w += acc.w;
  *((float4*)orow) = o;
}

extern "C" void kernel_launch(void* const* d_in, const int* in_sizes, int n_in,
                              void* d_out, int out_size, void* d_ws, size_t ws_size,
                              hipStream_t stream) {
  const float* feat_drug = (const float*)d_in[0];
  const float* feat_dis  = (const float*)d_in[1];
  const float* W_ind = (const float*)d_in[2];
  const float* b_ind = (const float*)d_in[3];
  const float* a_ind = (const float*)d_in[4];
  const float* W_rev = (const float*)d_in[5];
  const float* b_rev = (const float*)d_in[6];
  const float* a_rev = (const float*)d_in[7];
  const float* W_dd  = (const float*)d_in[8];
  const float* b_dd  = (const float*)d_in[9];
  const float* a_dd  = (const float*)d_in[10];
  const int* src_ind = (const int*)d_in[11];
  const int* dst_ind = (const int*)d_in[12];
  const int* src_rev = (const int*)d_in[13];
  const int* dst_rev = (const int*)d_in[14];
  const int* src_dd  = (const int*)d_in[15];
  const int* dst_dd  = (const int*)d_in[16];

  const int nDrug = in_sizes[0] / D;
  const int nDis  = in_sizes[1] / D;
  const int E     = in_sizes[11];

  // ---- workspace carve-up (~35 MB) ----
  float* ws = (float*)d_ws;
  float* Wh_ind = ws; ws += (size_t)nDrug * D;
  float* Wh_rev = ws; ws += (size_t)nDis  * D;
  float* Wh_dd  = ws; ws += (size_t)nDis  * D;
  float* el_ind = ws; ws += nDrug;   // drug  (src of ind)
  float* er_ind = ws; ws += nDis;    // dis   (dst key of ind)
  float* el_rev = ws; ws += nDis;    // dis   (src of rev)
  float* er_rev = ws; ws += nDrug;   // drug  (dst key of rev)
  float* el_dd  = ws; ws += nDis;
  float* er_dd  = ws; ws += nDis;
  unsigned int* cnt_ind = (unsigned int*)ws; ws += nDis;   // counters, contiguous
  unsigned int* cnt_rev = (unsigned int*)ws; ws += nDrug;
  unsigned int* cnt_dd  = (unsigned int*)ws; ws += nDis;
  unsigned int* cur_ind = (unsigned int*)ws; ws += nDis;
  unsigned int* cur_rev = (unsigned int*)ws; ws += nDrug;
  unsigned int* cur_dd  = (unsigned int*)ws; ws += nDis;
  unsigned int* rp_ind  = (unsigned int*)ws; ws += nDis  + 1;
  unsigned int* rp_rev  = (unsigned int*)ws; ws += nDrug + 1;
  unsigned int* rp_dd   = (unsigned int*)ws; ws += nDis  + 1;
  int* srcs_ind = (int*)ws; ws += E;
  int* srcs_rev = (int*)ws; ws += E;
  int* srcs_dd  = (int*)ws; ws += E;

  float* h_drug = (float*)d_out;
  float* h_dis  = h_drug + (size_t)nDrug * D;

  // ---- 1) init outputs + degree counters ----
  hrgcn_init_kernel<<<2048, 256, 0, stream>>>(
      (float*)d_out, out_size, cnt_ind, nDis + nDrug + nDis);

  // ---- 2) WMMA GEMMs: Wh = feat @ W + b ----
  hrgcn_gemm_bias_kernel<<<nDrug / 16, 256, 0, stream>>>(feat_drug, W_ind, b_ind, Wh_ind, nDrug);
  hrgcn_gemm_bias_kernel<<<nDis  / 16, 256, 0, stream>>>(feat_dis,  W_rev, b_rev, Wh_rev, nDis);
  hrgcn_gemm_bias_kernel<<<nDis  / 16, 256, 0, stream>>>(feat_dis,  W_dd,  b_dd,  Wh_dd,  nDis);

  // ---- 3) per-node attention scalars ----
  const int dotGridDrug = (nDrug + 7) / 8, dotGridDis = (nDis + 7) / 8;
  hrgcn_dot_a_kernel<<<dotGridDrug, 256, 0, stream>>>(Wh_ind, a_ind,     el_ind, nDrug);
  hrgcn_dot_a_kernel<<<dotGridDis,  256, 0, stream>>>(Wh_rev, a_ind + D, er_ind, nDis);
  hrgcn_dot_a_kernel<<<dotGridDis,  256, 0, stream>>>(Wh_rev, a_rev,     el_rev, nDis);
  hrgcn_dot_a_kernel<<<dotGridDrug, 256, 0, stream>>>(Wh_ind, a_rev + D, er_rev, nDrug);
  hrgcn_dot_a_kernel<<<dotGridDis,  256, 0, stream>>>(Wh_dd,  a_dd,      el_dd,  nDis);
  hrgcn_dot_a_kernel<<<dotGridDis,  256, 0, stream>>>(Wh_dd,  a_dd + D,  er_dd,  nDis);

  // ---- 4) CSR build (counting sort by dst) per etype ----
  const int eg = (E + 255) / 256;
  hrgcn_hist_kernel<<<eg, 256, 0, stream>>>(dst_ind, cnt_ind, E);
  hrgcn_hist_kernel<<<eg, 256, 0, stream>>>(dst_rev, cnt_rev, E);
  hrgcn_hist_kernel<<<eg, 256, 0, stream>>>(dst_dd,  cnt_dd,  E);

  hrgcn_scan_kernel<<<1, 1024, 0, stream>>>(cnt_ind, rp_ind, cur_ind, nDis);
  hrgcn_scan_kernel<<<1, 1024, 0, stream>>>(cnt_rev, rp_rev, cur_rev, nDrug);
  hrgcn_scan_kernel<<<1, 1024, 0, stream>>>(cnt_dd,  rp_dd,  cur_dd,  nDis);

  hrgcn_fill_kernel<<<eg, 256, 0, stream>>>(src_ind, dst_ind, cur_ind, srcs_ind, E);
  hrgcn_fill_kernel<<<eg, 256, 0, stream>>>(src_rev, dst_rev, cur_rev, srcs_rev, E);
  hrgcn_fill_kernel<<<eg, 256, 0, stream>>>(src_dd,  dst_dd,  cur_dd,  srcs_dd,  E);

  // ---- 5) atomic-free aggregation: one wave per destination node ----
  // indication: drug -> disease (into h_dis)
  hrgcn_agg_kernel<<<dotGridDis,  256, 0, stream>>>(rp_ind, srcs_ind, el_ind, er_ind, Wh_ind, h_dis,  nDis);
  // rev_indication: disease -> drug (into h_drug)
  hrgcn_agg_kernel<<<dotGridDrug, 256, 0, stream>>>(rp_rev, srcs_rev, el_rev, er_rev, Wh_rev, h_drug, nDrug);
  // disease_disease: disease -> disease (accumulates onto ind result)
  hrgcn_agg_kernel<<<dotGridDis,  256, 0, stream>>>(rp_dd,  srcs_dd,  el_dd,  er_dd,  Wh_dd,  h_dis,  nDis);
}